// S4DNeuroModel_75350906241027
// MI455X (gfx1250) — compile-verified
//
#include <hip/hip_runtime.h>
#include <hip/hip_bf16.h>
#include <math.h>

#define B_   32
#define L_   2048
#define IN_  182
#define OUT_ 2
#define H_   256
#define N2_  32
#define NL_  4

typedef __attribute__((ext_vector_type(2))) float v2f;
typedef __attribute__((ext_vector_type(8))) float v8f;

#define AS1 __attribute__((address_space(1)))
#define AS3 __attribute__((address_space(3)))

#if defined(__has_builtin)
#if __has_builtin(__builtin_amdgcn_global_load_async_to_lds_b32) && \
    __has_builtin(__builtin_amdgcn_s_wait_asynccnt)
#define USE_ASYNC_LDS 1
#endif
#endif

__device__ __forceinline__ float gelu_erf(float x) {
    return 0.5f * x * (1.0f + erff(x * 0.70710678118654752f));
}

__device__ __forceinline__ v8f wmma_f32(v2f a, v2f b, v8f c) {
    return __builtin_amdgcn_wmma_f32_16x16x4_f32(
        false, a, false, b, (short)0, c, false, false);
}

// ---------------------------------------------------------------------------
// Precompute per-layer SSM discretization:
//   lambda = exp(dt*A), Ceff = 2 * C * (lambda - 1) / A   (A = -exp(logAre) + i*Aim)
// ---------------------------------------------------------------------------
__global__ void precompute_kernel(
    const float* __restrict__ C2, const float* __restrict__ log_dt,
    const float* __restrict__ log_A_real, const float* __restrict__ A_imag,
    float* __restrict__ lam_r, float* __restrict__ lam_i,
    float* __restrict__ ce_r,  float* __restrict__ ce_i)
{
    int idx = blockIdx.x * blockDim.x + threadIdx.x;
    if (idx >= NL_ * H_ * N2_) return;
    int lh = idx / N2_;                         // layer*H + h
    float dt = __expf(log_dt[lh]);
    float ar = -__expf(log_A_real[idx]);
    float ai = A_imag[idx];
    float dtar = ar * dt, dtai = ai * dt;
    float e  = __expf(dtar);
    float lr = e * __cosf(dtai);
    float li = e * __sinf(dtai);
    lam_r[idx] = lr; lam_i[idx] = li;
    float e1r = lr - 1.0f, e1i = li;            // lambda - 1
    float c2r = C2[idx * 2 + 0], c2i = C2[idx * 2 + 1];
    float nr = c2r * e1r - c2i * e1i;           // C * (lambda-1)
    float ni = c2r * e1i + c2i * e1r;
    float inv = 2.0f / (ar * ar + ai * ai);     // fold the 2.0 from 2*Re(...)
    ce_r[idx] = (nr * ar + ni * ai) * inv;      // num * conj(A) / |A|^2
    ce_i[idx] = (ni * ar - nr * ai) * inv;
}

// ---------------------------------------------------------------------------
// Input projection: x[b][h][l] = sum_i in[b][l][i]*Win[i][h] + b_in[h]
// WMMA f32 16x16x4: A = Win^T tile (M=h, K=i), B = in^T tile (K=i, N=l)
// ---------------------------------------------------------------------------
#define KP 188   // LDS row stride (words), IN_ zero-padded to >=184

__global__ __launch_bounds__(512) void input_proj_kernel(
    const float* __restrict__ in, const float* __restrict__ Win,
    const float* __restrict__ b_in, float* __restrict__ x)
{
    __shared__ float s_in[16 * KP];
    const int nlt = L_ / 16;
    int b  = blockIdx.x / nlt;
    int l0 = (blockIdx.x % nlt) * 16;
    int tid = threadIdx.x;

    // stage in[b][l0+j][i] -> s_in[j*KP + i], zero-pad past IN_
    {
        int j  = tid >> 5;           // 0..15
        int ib = tid & 31;
        const float* src = in + ((size_t)b * L_ + (l0 + j)) * IN_;
#ifdef USE_ASYNC_LDS
        for (int r = 0; r < 6; ++r) {
            int i = ib + 32 * r;
            if (i < IN_) {
                __builtin_amdgcn_global_load_async_to_lds_b32(
                    (AS1 int*)(src + i), (AS3 int*)&s_in[j * KP + i], 0, 0);
            } else if (i < KP) {
                s_in[j * KP + i] = 0.0f;
            }
        }
        __builtin_amdgcn_s_wait_asynccnt(0);
#else
        for (int r = 0; r < 6; ++r) {
            int i = ib + 32 * r;
            if (i < KP) s_in[j * KP + i] = (i < IN_) ? src[i] : 0.0f;
        }
#endif
    }
    __syncthreads();

    int wave = tid >> 5;             // 0..15 -> h-tile
    int lane = tid & 31;
    int m  = lane & 15;              // A row (h offset) == B col (l offset)
    int hi = lane >> 4;              // lane half selects K pair
    int h0 = wave * 16;
    const float* winp = Win + h0 + m;      // column base, row stride H_

    // software-pipelined K loop over 46 tiles (k0 = 0..180); only the last
    // tile needs bounds handling on the A (Win) side.
    v8f acc = {};
    v2f a_cur, b_cur;
    {
        int kb = 2 * hi;
        a_cur.x = winp[(size_t)kb * H_];
        a_cur.y = winp[(size_t)(kb + 1) * H_];
        b_cur = *(const v2f*)&s_in[m * KP + kb];
    }
    for (int k0 = 0; k0 < 176; k0 += 4) {          // preload k0+4 (kb<=179, in bounds)
        int kb = k0 + 4 + 2 * hi;
        v2f a_n, b_n;
        a_n.x = winp[(size_t)kb * H_];
        a_n.y = winp[(size_t)(kb + 1) * H_];
        b_n = *(const v2f*)&s_in[m * KP + kb];
        acc = wmma_f32(a_cur, b_cur, acc);
        a_cur = a_n; b_cur = b_n;
    }
    {   // tail tile (k0 = 180): kb in {180,182}; guard via cndmask (no exec branch)
        int kb  = 180 + 2 * hi;
        int kc0 = (kb     < IN_) ? kb     : 0;
        int kc1 = (kb + 1 < IN_) ? kb + 1 : 0;
        v2f a_n, b_n;
        a_n.x = winp[(size_t)kc0 * H_];
        a_n.y = winp[(size_t)kc1 * H_];
        a_n.x = (kb     < IN_) ? a_n.x : 0.0f;
        a_n.y = (kb + 1 < IN_) ? a_n.y : 0.0f;
        b_n = *(const v2f*)&s_in[m * KP + kb];      // LDS zero-padded
        acc = wmma_f32(a_cur, b_cur, acc);
        acc = wmma_f32(a_n, b_n, acc);
    }
    // D VGPR j: rows j (lanes 0-15), j+8 (lanes 16-31); col = m
    for (int j = 0; j < 8; ++j) {
        int row = h0 + j + 8 * hi;
        x[((size_t)b * H_ + row) * L_ + l0 + m] = acc[j] + b_in[row];
    }
}

// ---------------------------------------------------------------------------
// S4D diagonal recurrence (== FFT conv of the reference), fused +D*u and GELU.
// One wave per (b,h); lane = state index n (N2 == 32 == wave32).
// ---------------------------------------------------------------------------
__global__ __launch_bounds__(256) void s4d_recur_kernel(
    const float* __restrict__ x,
    const float* __restrict__ lam_r, const float* __restrict__ lam_i,
    const float* __restrict__ ce_r,  const float* __restrict__ ce_i,
    const float* __restrict__ Dp, float* __restrict__ y, int layer)
{
    int wgl  = blockIdx.x * (blockDim.x >> 5) + (threadIdx.x >> 5);
    int lane = threadIdx.x & 31;
    int b = wgl / H_;
    int h = wgl % H_;
    int pidx = (layer * H_ + h) * N2_ + lane;
    float lr = lam_r[pidx], li = lam_i[pidx];
    float cr = ce_r[pidx],  ci = ce_i[pidx];
    float d  = Dp[layer * H_ + h];
    const float* u_ptr = x + ((size_t)b * H_ + h) * L_;
    float*       y_ptr = y + ((size_t)b * H_ + h) * L_;

    float sr = 0.0f, si = 0.0f, myval = 0.0f;
    for (int l = 0; l < L_; ++l) {
        float u   = u_ptr[l];                        // uniform broadcast load
        float nsr = fmaf(lr, sr, fmaf(-li, si, u));  // s = lambda*s + u
        float nsi = fmaf(lr, si, li * sr);
        sr = nsr; si = nsi;
        float t = fmaf(cr, sr, -ci * si);            // Re(Ceff * s), per n
        t += __shfl_xor(t, 16, 32);                  // butterfly sum over n
        t += __shfl_xor(t,  8, 32);
        t += __shfl_xor(t,  4, 32);
        t += __shfl_xor(t,  2, 32);
        t += __shfl_xor(t,  1, 32);
        float val = gelu_erf(t + d * u);
        if ((l & 31) == lane) myval = val;           // park in owning lane
        if ((l & 31) == 31)  y_ptr[(l & ~31) + lane] = myval;  // coalesced
    }
}

// ---------------------------------------------------------------------------
// GLU 1x1 conv: z = Wg(2H x H) @ y + bg ; out = z[0:H] * sigmoid(z[H:2H]).
// Workgroup = 16 waves per (b, 16-wide l tile); wave w owns rows [16w,16w+16)
// of both the 'a' half and the 'g' half (shared B operand, fused GLU).
// ---------------------------------------------------------------------------
#define HP 260   // LDS row stride (words) for the 16 x H y-tile (transposed)

__global__ __launch_bounds__(512) void glu_gemm_kernel(
    const float* __restrict__ y, const float* __restrict__ Wg,
    const float* __restrict__ bg, float* __restrict__ xo, int layer)
{
    __shared__ float s_y[16 * HP];
    const int nlt = L_ / 16;
    int b  = blockIdx.x / nlt;
    int l0 = (blockIdx.x % nlt) * 16;
    int tid = threadIdx.x;

    // stage y[b][h][l0+j] -> s_y[j*HP + h] (transpose; coalesced global reads)
    {
        int j  = tid & 15;
        int hb = tid >> 4;               // 0..31
#ifdef USE_ASYNC_LDS
        for (int i = 0; i < 8; ++i) {
            int h = hb + 32 * i;
            const float* gp = &y[((size_t)b * H_ + h) * L_ + l0 + j];
            __builtin_amdgcn_global_load_async_to_lds_b32(
                (AS1 int*)gp, (AS3 int*)&s_y[j * HP + h], 0, 0);
        }
        __builtin_amdgcn_s_wait_asynccnt(0);
#else
        for (int i = 0; i < 8; ++i) {
            int h = hb + 32 * i;
            s_y[j * HP + h] = y[((size_t)b * H_ + h) * L_ + l0 + j];
        }
#endif
    }
    __syncthreads();

    int wave = tid >> 5;                 // 0..15
    int lane = tid & 31;
    int m  = lane & 15;
    int hi = lane >> 4;
    int m0 = wave * 16;
    const float* Wa = Wg + (size_t)layer * 2 * H_ * H_ + (size_t)(m0 + m) * H_;
    const float* Wb = Wa + (size_t)H_ * H_;
    const float* bl = bg + (size_t)layer * 2 * H_;

    v8f acc_a = {}, acc_g = {};
    v2f a1c, a2c, bc;
    {
        int kb = 2 * hi;
        a1c = *(const v2f*)&Wa[kb];
        a2c = *(const v2f*)&Wb[kb];
        bc  = *(const v2f*)&s_y[m * HP + kb];
    }
    for (int k0 = 0; k0 < H_ - 4; k0 += 4) {       // preload next tile, 2 WMMAs
        int kb = k0 + 4 + 2 * hi;
        v2f a1n = *(const v2f*)&Wa[kb];
        v2f a2n = *(const v2f*)&Wb[kb];
        v2f bn  = *(const v2f*)&s_y[m * HP + kb];
        acc_a = wmma_f32(a1c, bc, acc_a);
        acc_g = wmma_f32(a2c, bc, acc_g);
        a1c = a1n; a2c = a2n; bc = bn;
    }
    acc_a = wmma_f32(a1c, bc, acc_a);
    acc_g = wmma_f32(a2c, bc, acc_g);

    for (int j = 0; j < 8; ++j) {
        int row = m0 + j + 8 * hi;
        float av = acc_a[j] + bl[row];
        float gv = acc_g[j] + bl[row + H_];
        float o  = av * (1.0f / (1.0f + __expf(-gv)));
        xo[((size_t)b * H_ + row) * L_ + l0 + m] = o;
    }
}

// ---------------------------------------------------------------------------
// Output projection: out[b][l][o] = sum_h x[b][h][l]*Wout[h][o] + b_out[o]
// ---------------------------------------------------------------------------
__global__ __launch_bounds__(256) void out_proj_kernel(
    const float* __restrict__ x, const float* __restrict__ Wout,
    const float* __restrict__ b_out, float* __restrict__ out)
{
    int idx = blockIdx.x * blockDim.x + threadIdx.x;   // over B*L
    int b = idx / L_;
    int l = idx % L_;
    float a0 = b_out[0], a1 = b_out[1];
    const float* xp = x + (size_t)b * H_ * L_ + l;
    for (int h = 0; h < H_; ++h) {
        float u = xp[(size_t)h * L_];                  // lanes over l: coalesced
        a0 = fmaf(u, Wout[h * 2 + 0], a0);
        a1 = fmaf(u, Wout[h * 2 + 1], a1);
    }
    out[(size_t)idx * 2 + 0] = a0;
    out[(size_t)idx * 2 + 1] = a1;
}

// ---------------------------------------------------------------------------
extern "C" void kernel_launch(void* const* d_in, const int* in_sizes, int n_in,
                              void* d_out, int out_size, void* d_ws, size_t ws_size,
                              hipStream_t stream) {
    const float* neural_input = (const float*)d_in[0];
    const float* Win        = (const float*)d_in[1];
    const float* b_in       = (const float*)d_in[2];
    const float* C2         = (const float*)d_in[3];
    const float* log_dt     = (const float*)d_in[4];
    const float* log_A_real = (const float*)d_in[5];
    const float* A_imag     = (const float*)d_in[6];
    const float* Dp         = (const float*)d_in[7];
    const float* Wglu       = (const float*)d_in[8];
    const float* bglu       = (const float*)d_in[9];
    const float* Wout       = (const float*)d_in[10];
    const float* b_out      = (const float*)d_in[11];

    const size_t act = (size_t)B_ * H_ * L_;
    float* xbuf  = (float*)d_ws;
    float* ybuf  = xbuf + act;
    float* lam_r = ybuf + act;
    float* lam_i = lam_r + NL_ * H_ * N2_;
    float* ce_r  = lam_i + NL_ * H_ * N2_;
    float* ce_i  = ce_r  + NL_ * H_ * N2_;

    precompute_kernel<<<(NL_ * H_ * N2_ + 255) / 256, 256, 0, stream>>>(
        C2, log_dt, log_A_real, A_imag, lam_r, lam_i, ce_r, ce_i);

    input_proj_kernel<<<B_ * (L_ / 16), 512, 0, stream>>>(
        neural_input, Win, b_in, xbuf);

    for (int layer = 0; layer < NL_; ++layer) {
        s4d_recur_kernel<<<(B_ * H_) / 8, 256, 0, stream>>>(
            xbuf, lam_r, lam_i, ce_r, ce_i, Dp, ybuf, layer);
        glu_gemm_kernel<<<B_ * (L_ / 16), 512, 0, stream>>>(
            ybuf, Wglu, bglu, xbuf, layer);
    }

    out_proj_kernel<<<(B_ * L_) / 256, 256, 0, stream>>>(
        xbuf, Wout, b_out, (float*)d_out);
}